// Unpool_36988258353720
// MI455X (gfx1250) — compile-verified
//
#include <hip/hip_runtime.h>

typedef __attribute__((ext_vector_type(2))) float v2f;
typedef __attribute__((ext_vector_type(8))) float v8f;

#define DIM 128
#define LN_EPS 1e-5f
#define TILES 8   // tile iterations per wave in the fused kernel

// ---------------- init workspace ----------------
__global__ void k_init(float* __restrict__ sums, float* __restrict__ cnt,
                       int* __restrict__ winner, int N, int Esmall) {
  long long t = (long long)blockIdx.x * blockDim.x + threadIdx.x;
  long long total = (long long)N * DIM;
  if (t < total) sums[t] = 0.0f;
  if (t < N) cnt[t] = 0.0f;
  if (t < Esmall) winner[t] = -1;
}

// ---------------- x_new = x[comp] ----------------
__global__ void k_gather_x(const float* __restrict__ x, const int* __restrict__ comp,
                           float* __restrict__ xn, int N) {
  long long t = (long long)blockIdx.x * blockDim.x + threadIdx.x;
  if (t >= (long long)N * 32) return;
  int i = (int)(t >> 5);
  int d0 = ((int)t & 31) * 4;
  float4 v = *(const float4*)(x + (size_t)comp[i] * DIM + d0);
  *(float4*)(xn + (size_t)i * DIM + d0) = v;
}

// ---------------- winner[c_perm[j]] = max j (deterministic "any" scatter) -------
__global__ void k_winner(const int* __restrict__ c_perm, int* __restrict__ winner, int Epre) {
  int j = blockIdx.x * blockDim.x + threadIdx.x;
  if (j < Epre) atomicMax(&winner[c_perm[j]], j);
}

// ---------------- segment sum over dst + counts ----------------
__global__ void k_scatter(const int* __restrict__ ei, const int* __restrict__ winner,
                          const int* __restrict__ idx, const float* __restrict__ ea0,
                          float* __restrict__ sums, float* __restrict__ cnt,
                          int E, int Esmall) {
  long long t = (long long)blockIdx.x * blockDim.x + threadIdx.x;
  if (t >= (long long)E * 32) return;
  int e = (int)(t >> 5);
  int lane = (int)t & 31;
  int dstv = ei[(size_t)E + e];
  if (lane == 0) atomicAdd(&cnt[dstv], 1.0f);
  if (e < Esmall) {
    int wn = winner[e];
    if (wn >= 0) {
      int d0 = lane * 4;
      float4 v = *(const float4*)(ea0 + (size_t)idx[wn] * DIM + d0);
      float* sp = sums + (size_t)dstv * DIM + d0;
      atomicAdd(sp + 0, v.x);
      atomicAdd(sp + 1, v.y);
      atomicAdd(sp + 2, v.z);
      atomicAdd(sp + 3, v.w);
    }
  }
}

// ---------------- fused: build h, WMMA GEMM (f32 16x16x4), bias+LN+ReLU -------
// block = 128 threads = 4 waves; each wave processes TILES x 16-edge tiles.
// W is staged once per block into LDS, pre-swizzled to B-fragment order.
__global__ __launch_bounds__(128)
void k_fused(const int* __restrict__ ei, const int* __restrict__ winner,
             const int* __restrict__ idx, const float* __restrict__ ea0,
             const float* __restrict__ sums, const float* __restrict__ cnt,
             const float* __restrict__ xn, const float* __restrict__ Wm,
             const float* __restrict__ bias, const float* __restrict__ gamma,
             const float* __restrict__ beta, float* __restrict__ outE,
             int E, int Esmall) {
  constexpr int RS = 132;                 // padded row stride (bank-conflict free)
  __shared__ float WB[16384];             // 64 KB: W in B-fragment order
  __shared__ float hs[4][16][RS];         // 33,792 B: per-wave h tiles

  const int wv   = threadIdx.x >> 5;
  const int lane = threadIdx.x & 31;
  const int half = lane >> 4;             // 0: lanes 0-15, 1: lanes 16-31
  const int l16  = lane & 15;

  // ---- stage W -> LDS in fragment order (coalesced float4 global reads) ----
  // WB dword layout: [(k4*8 + nt)*64 + lane*2 + j] = W[nt*16+l16][k4*4 + 2*half + j]
  for (int i = threadIdx.x; i < 4096; i += 128) {
    int o  = i >> 5;            // output feature (row of W)
    int dq = (i & 31) * 4;      // input feature (col of W), multiple of 4
    float4 w = *(const float4*)(Wm + (size_t)o * DIM + dq);
    int k4  = dq >> 2;
    int l16o = o & 15;
    int nt  = o >> 4;
    int base = (k4 * 8 + nt) * 64;
    *(float2*)&WB[base + l16o * 2]        = make_float2(w.x, w.y);  // half 0
    *(float2*)&WB[base + (16 + l16o) * 2] = make_float2(w.z, w.w);  // half 1
  }
  __syncthreads();

  for (int it = 0; it < TILES; ++it) {
    const long long e0 = (((long long)blockIdx.x * TILES + it) * 4 + wv) * 16;

    // ---- phase 1: h tile -> LDS ----
    const int d0 = lane * 4;
    for (int i = 0; i < 16; ++i) {
      long long e = e0 + i;
      float4 h4 = make_float4(0.f, 0.f, 0.f, 0.f);
      if (e < E) {
        int src = ei[e];
        float4 ea = make_float4(0.f, 0.f, 0.f, 0.f);
        if (e < Esmall) {
          int wn = winner[e];
          if (wn >= 0) ea = *(const float4*)(ea0 + (size_t)idx[wn] * DIM + d0);
        }
        float ic = 1.0f / fmaxf(cnt[src], 1.0f);
        float4 sm = *(const float4*)(sums + (size_t)src * DIM + d0);
        float4 xv = *(const float4*)(xn + (size_t)src * DIM + d0);
        const float third = 1.0f / 3.0f;
        h4.x = (ea.x + sm.x * ic + xv.x) * third;
        h4.y = (ea.y + sm.y * ic + xv.y) * third;
        h4.z = (ea.z + sm.z * ic + xv.z) * third;
        h4.w = (ea.w + sm.w * ic + xv.w) * third;
      }
      hs[wv][i][d0 + 0] = h4.x;
      hs[wv][i][d0 + 1] = h4.y;
      hs[wv][i][d0 + 2] = h4.z;
      hs[wv][i][d0 + 3] = h4.w;
    }
    __syncthreads();

    // ---- phase 2: out[m][n] = sum_k h[m][k] * W[n][k] via V_WMMA_F32_16X16X4_F32
    // A 16x4 f32: lane l16 = row M, half selects K pair {k0+2h, k0+2h+1}
    // B 4x16  f32: conflict-free ds_load_b64 from pre-swizzled WB
    v8f acc[8] = {};
    for (int k4 = 0; k4 < 32; ++k4) {
      int ka = k4 * 4 + 2 * half;
      v2f a;
      a[0] = hs[wv][l16][ka];
      a[1] = hs[wv][l16][ka + 1];
      const float* wb = &WB[k4 * 512 + lane * 2];
#pragma unroll
      for (int nt = 0; nt < 8; ++nt) {
        v2f bf = *(const v2f*)(wb + nt * 64);
        acc[nt] = __builtin_amdgcn_wmma_f32_16x16x4_f32(
            false, a, false, bf, (short)0, acc[nt], false, false);
      }
    }

    // ---- phase 3: bias + LayerNorm + ReLU in C-fragment layout ----
    // C layout: acc[nt][r] @ lane = elem (m = r + 8*half, n = nt*16 + l16)
#pragma unroll
    for (int nt = 0; nt < 8; ++nt) {
      float bb = bias[nt * 16 + l16];
#pragma unroll
      for (int r = 0; r < 8; ++r) acc[nt][r] += bb;
    }

    float s[8], q[8];
#pragma unroll
    for (int r = 0; r < 8; ++r) { s[r] = 0.f; q[r] = 0.f; }
#pragma unroll
    for (int nt = 0; nt < 8; ++nt)
#pragma unroll
      for (int r = 0; r < 8; ++r) {
        float v = acc[nt][r];
        s[r] += v;
        q[r] += v * v;
      }
    // butterfly within each 16-lane half (masks < 16 stay inside the half)
#pragma unroll
    for (int off = 1; off < 16; off <<= 1) {
#pragma unroll
      for (int r = 0; r < 8; ++r) {
        s[r] += __shfl_xor(s[r], off, 32);
        q[r] += __shfl_xor(q[r], off, 32);
      }
    }

#pragma unroll
    for (int r = 0; r < 8; ++r) {
      int m = r + 8 * half;
      long long e = e0 + m;
      float mu = s[r] * (1.0f / DIM);
      float var = q[r] * (1.0f / DIM) - mu * mu;
      float rs = rsqrtf(var + LN_EPS);
      if (e < E) {
#pragma unroll
        for (int nt = 0; nt < 8; ++nt) {
          int n = nt * 16 + l16;
          float v = (acc[nt][r] - mu) * rs * gamma[n] + beta[n];
          outE[(size_t)e * DIM + n] = fmaxf(v, 0.0f);
        }
      }
    }
    __syncthreads();
  }
}

extern "C" void kernel_launch(void* const* d_in, const int* in_sizes, int n_in,
                              void* d_out, int out_size, void* d_ws, size_t ws_size,
                              hipStream_t stream) {
  const float* x      = (const float*)d_in[0];
  const int*   comp   = (const int*)d_in[1];
  const int*   ei     = (const int*)d_in[2];
  const float* ea0    = (const float*)d_in[3];
  const int*   idx    = (const int*)d_in[4];
  const int*   c_perm = (const int*)d_in[5];
  /* d_in[6] = mask_idx = arange -> identity prefix, unused */
  const float* W      = (const float*)d_in[7];
  const float* b      = (const float*)d_in[8];
  const float* gamma  = (const float*)d_in[9];
  const float* beta   = (const float*)d_in[10];

  const int N      = in_sizes[1];
  const int E      = in_sizes[2] / 2;
  const int Epre   = in_sizes[4];
  const int Esmall = in_sizes[6];

  // workspace layout (256B aligned): winner[Esmall] | cnt[N] | sums[N*128]
  size_t off = 0;
  int* winner = (int*)((char*)d_ws + off);
  off += (size_t)Esmall * 4; off = (off + 255) & ~(size_t)255;
  float* cnt = (float*)((char*)d_ws + off);
  off += (size_t)N * 4; off = (off + 255) & ~(size_t)255;
  float* sums = (float*)((char*)d_ws + off);

  float* xn   = (float*)d_out;                    // output 0: x_new [N,128]
  float* outE = (float*)d_out + (size_t)N * DIM;  // output 1: edge_attr_out [E,128]

  long long initT = (long long)N * DIM;
  k_init<<<(int)((initT + 255) / 256), 256, 0, stream>>>(sums, cnt, winner, N, Esmall);

  long long gxT = (long long)N * 32;
  k_gather_x<<<(int)((gxT + 255) / 256), 256, 0, stream>>>(x, comp, xn, N);

  k_winner<<<(Epre + 255) / 256, 256, 0, stream>>>(c_perm, winner, Epre);

  long long scT = (long long)E * 32;
  k_scatter<<<(int)((scT + 255) / 256), 256, 0, stream>>>(ei, winner, idx, ea0,
                                                          sums, cnt, E, Esmall);

  // 4 waves x TILES x 16 edges = 512 edges per block
  int edgesPerBlock = 4 * TILES * 16;
  int fblocks = (E + edgesPerBlock - 1) / edgesPerBlock;
  k_fused<<<fblocks, 128, 0, stream>>>(ei, winner, idx, ea0, sums, cnt, xn,
                                       W, b, gamma, beta, outE, E, Esmall);
}